// PC_Encoder_5454608466698
// MI455X (gfx1250) — compile-verified
//
#include <hip/hip_runtime.h>
#include <hip/hip_bf16.h>

typedef __attribute__((ext_vector_type(2))) float v2f;
typedef __attribute__((ext_vector_type(8))) float v8f;

#define BATCH 8
#define NPTS  8192
#define KNN_K 32
#define FPS_THREADS 1024
#define FPS_WAVES   (FPS_THREADS / 32)
#define MLP_THREADS 128   // 4 wave32 waves

// (value, index) argmax combine with reference tie-break: larger value wins,
// equal value -> smaller index wins (jnp.argmax picks first occurrence).
__device__ __forceinline__ void argmax_combine(float& v, int& i, float ov, int oi) {
  if (ov > v || (ov == v && oi < i)) { v = ov; i = oi; }
}

// Wave32 reduction, no barriers: after this, lane 0 holds the wave's argmax.
__device__ __forceinline__ void wave_argmax(float& v, int& i) {
#pragma unroll
  for (int o = 16; o > 0; o >>= 1) {
    const float ov = __shfl_down(v, o, 32);
    const int   oi = __shfl_down(i, o, 32);
    argmax_combine(v, i, ov, oi);
  }
}

// ---------------------------------------------------------------------------
// Farthest point sampling: one workgroup per batch element. mind[] in LDS.
// Sequential argmax chain is the whole kernel's critical path, so the per-step
// reduction uses wave shuffles (2 barriers/step instead of a 10-barrier tree).
// ---------------------------------------------------------------------------
__global__ void __launch_bounds__(FPS_THREADS)
fps_kernel(const float* __restrict__ pos, int* __restrict__ idxOut,
           float* __restrict__ posOut, int N, int m) {
  __shared__ float mind[NPTS];
  __shared__ float redv[FPS_WAVES];
  __shared__ int   redi[FPS_WAVES];
  __shared__ int   s_last;
  const int b = blockIdx.x;
  const int tid = threadIdx.x;
  const int lane = tid & 31, wave = tid >> 5;
  const float* P = pos + (size_t)b * N * 3;
  int*   idxB = idxOut + (size_t)b * m;
  float* posB = posOut + (size_t)b * m * 3;

  for (int j = tid; j < N; j += FPS_THREADS) mind[j] = 1e10f;
  if (tid == 0) {
    idxB[0] = 0;
    posB[0] = P[0]; posB[1] = P[1]; posB[2] = P[2];
    s_last = 0;
  }
  __syncthreads();
  int last = 0;
  for (int i = 1; i < m; ++i) {
    const float lx = P[last * 3 + 0], ly = P[last * 3 + 1], lz = P[last * 3 + 2];
    float bestv = -1.0f; int besti = N;
    for (int j = tid; j < N; j += FPS_THREADS) {
      const float dx = P[j * 3 + 0] - lx;
      const float dy = P[j * 3 + 1] - ly;
      const float dz = P[j * 3 + 2] - lz;
      float mn = fminf(mind[j], dx * dx + dy * dy + dz * dz);
      mind[j] = mn;
      if (mn > bestv) { bestv = mn; besti = j; }  // strict > keeps first index
    }
    wave_argmax(bestv, besti);                  // no barriers
    if (lane == 0) { redv[wave] = bestv; redi[wave] = besti; }
    __syncthreads();
    if (wave == 0) {
      float v = redv[lane]; int ix = redi[lane];   // FPS_WAVES == 32
      wave_argmax(v, ix);
      if (lane == 0) {
        s_last = ix;
        idxB[i] = ix;
        posB[i * 3 + 0] = P[ix * 3 + 0];
        posB[i * 3 + 1] = P[ix * 3 + 1];
        posB[i * 3 + 2] = P[ix * 3 + 2];
      }
    }
    __syncthreads();
    last = s_last;
  }
}

// ---------------------------------------------------------------------------
// Brute-force KNN: one thread per query, register-resident top-32 via
// replace-worst (unrolled so arrays stay in VGPRs). Only the SET of 32
// nearest matters (max-pool downstream is order-invariant).
// ---------------------------------------------------------------------------
__global__ void knn_kernel(const float* __restrict__ pos, const float* __restrict__ posQ,
                           int* __restrict__ nbr, int N, int M, int total) {
  const int g = blockIdx.x * blockDim.x + threadIdx.x;
  if (g >= total) return;
  const int b = g / M, q = g % M;
  const float* P = pos + (size_t)b * N * 3;
  const float qx = posQ[((size_t)b * M + q) * 3 + 0];
  const float qy = posQ[((size_t)b * M + q) * 3 + 1];
  const float qz = posQ[((size_t)b * M + q) * 3 + 2];

  float bd[KNN_K]; int bi[KNN_K];
#pragma unroll
  for (int t = 0; t < KNN_K; ++t) { bd[t] = 3.4e38f; bi[t] = -1; }
  float wmax = 3.4e38f; int wslot = 0;

  for (int j = 0; j < N; ++j) {
    const float dx = P[j * 3 + 0] - qx;
    const float dy = P[j * 3 + 1] - qy;
    const float dz = P[j * 3 + 2] - qz;
    const float d = dx * dx + dy * dy + dz * dz;
    if (d < wmax) {
#pragma unroll
      for (int t = 0; t < KNN_K; ++t) if (t == wslot) { bd[t] = d; bi[t] = j; }
      float nm = -1.0f; int ns = 0;
#pragma unroll
      for (int t = 0; t < KNN_K; ++t) { if (bd[t] > nm) { nm = bd[t]; ns = t; } }
      wmax = nm; wslot = ns;
    }
  }
  int* nb = nbr + ((size_t)b * M + q) * KNN_K;
#pragma unroll
  for (int t = 0; t < KNN_K; ++t) nb[t] = bi[t];
}

// ---------------------------------------------------------------------------
// One 16x16 output tile per wave via V_WMMA_F32_16X16X4_F32.
// A (16x4 f32) lane layout (ISA 7.12.2): lane L: M=L&15, v0=K(koff), v1=K(koff+1)
// with koff = 2*(L>>4). B (4x16) mirrored on N. D: VGPR j -> M=j+8*(L>>4), N=L&15.
// ---------------------------------------------------------------------------
__device__ __forceinline__ void mlp_layer(const float* __restrict__ in, int Din,
                                          float* __restrict__ out, int Dout,
                                          const float* __restrict__ W,
                                          const float* __restrict__ bias,
                                          bool relu, int lane, int wave) {
  const int nct = Dout >> 4;          // 16-wide column tiles
  const int ntiles = 2 * nct;         // 2 row tiles (32 rows)
  const int koff = (lane >> 4) << 1;  // 0 or 2
  const int ln = lane & 15;
  const int half = (lane >> 4) * 8;
  for (int t = wave; t < ntiles; t += MLP_THREADS / 32) {   // wave-uniform
    const int rt = t / nct, ct = t % nct;
    const int rbase = rt * 16;
    v8f acc = {0.f, 0.f, 0.f, 0.f, 0.f, 0.f, 0.f, 0.f};
    for (int k0 = 0; k0 < Din; k0 += 4) {
      v2f a, bb;
      a.x  = in[(rbase + ln) * Din + k0 + koff];
      a.y  = in[(rbase + ln) * Din + k0 + koff + 1];
      bb.x = W[(k0 + koff) * Dout + ct * 16 + ln];
      bb.y = W[(k0 + koff + 1) * Dout + ct * 16 + ln];
      acc = __builtin_amdgcn_wmma_f32_16x16x4_f32(false, a, false, bb,
                                                  (short)0, acc, false, false);
    }
#pragma unroll
    for (int j = 0; j < 8; ++j) {
      const int row = rbase + j + half;
      const int col = ct * 16 + ln;
      float v = acc[j] + bias[col];
      if (relu) v = fmaxf(v, 0.0f);
      out[row * Dout + col] = v;
    }
  }
}

// ---------------------------------------------------------------------------
// Fused SA-module MLP: one workgroup per (batch, query). Gather 32 neighbor
// feature rows [x[nbr], rel] into LDS (K-padded), 3 WMMA layers ping-ponging
// between two LDS buffers, then max-pool over the 32 neighbors.
// LDS: 32*132*4 + 32*256*4 = ~49.7 KB per WG (of 320 KB/WGP).
// ---------------------------------------------------------------------------
__global__ void __launch_bounds__(MLP_THREADS)
sa_mlp_kernel(const float* __restrict__ xin, int Cin,
              const float* __restrict__ pos,   // [B,N,3] source positions
              const float* __restrict__ posQ,  // [B,M,3] sampled positions
              const int* __restrict__ nbr,     // [B,M,32]
              const float* __restrict__ W1, const float* __restrict__ b1, int D0, int D1,
              const float* __restrict__ W2, const float* __restrict__ b2, int D2,
              const float* __restrict__ W3, const float* __restrict__ b3, int D3,
              float* __restrict__ xout, int N, int M) {
  __shared__ float bufA[32 * 132];   // feat (<=132) and layer-2 out (<=128)
  __shared__ float bufB[32 * 256];   // layer-1 out (<=128) and layer-3 out (<=256)

  const int b = blockIdx.x / M;
  const int q = blockIdx.x % M;
  const int tid = threadIdx.x;
  const int lane = tid & 31, wave = tid >> 5;

  const int* nb = nbr + ((size_t)b * M + q) * KNN_K;
  const float* Pb = pos + (size_t)b * N * 3;
  const float* Xb = xin + (size_t)b * N * Cin;
  const float qx = posQ[((size_t)b * M + q) * 3 + 0];
  const float qy = posQ[((size_t)b * M + q) * 3 + 1];
  const float qz = posQ[((size_t)b * M + q) * 3 + 2];

  // Gather [x[nbr] | rel | zero-pad] into bufA, row-major stride D0.
  for (int e = tid; e < 32 * D0; e += MLP_THREADS) {
    const int r = e / D0, c = e % D0;
    const int n = nb[r];
    float v;
    if (c < Cin)           v = Xb[(size_t)n * Cin + c];
    else if (c == Cin)     v = Pb[n * 3 + 0] - qx;
    else if (c == Cin + 1) v = Pb[n * 3 + 1] - qy;
    else if (c == Cin + 2) v = Pb[n * 3 + 2] - qz;
    else                   v = 0.0f;
    bufA[r * D0 + c] = v;
  }
  __syncthreads();

  mlp_layer(bufA, D0, bufB, D1, W1, b1, true, lane, wave);
  __syncthreads();
  mlp_layer(bufB, D1, bufA, D2, W2, b2, true, lane, wave);
  __syncthreads();
  mlp_layer(bufA, D2, bufB, D3, W3, b3, false, lane, wave);  // plain_last
  __syncthreads();

  // Max over 32 neighbors per output channel.
  float* o = xout + ((size_t)b * M + q) * D3;
  for (int c = tid; c < D3; c += MLP_THREADS) {
    float mx = bufB[c];
#pragma unroll 4
    for (int r = 1; r < 32; ++r) mx = fmaxf(mx, bufB[r * D3 + c]);
    o[c] = mx;
  }
}

// Zero-pad weights along K to a multiple of 4 (WMMA f32 K=4 steps).
__global__ void pad_weight_kernel(const float* __restrict__ src, float* __restrict__ dst,
                                  int Kin, int Kpad, int Dout) {
  const int i = blockIdx.x * blockDim.x + threadIdx.x;
  if (i >= Kpad * Dout) return;
  const int k = i / Dout;
  dst[i] = (k < Kin) ? src[i] : 0.0f;
}

// xyz3 + batch3 into d_out tail.
__global__ void finalize_kernel(const float* __restrict__ pos3, float* __restrict__ out_xyz,
                                float* __restrict__ out_batch, int M) {
  const int i = blockIdx.x * blockDim.x + threadIdx.x;
  if (i < BATCH * M * 3) out_xyz[i] = pos3[i];
  if (i < BATCH * M)     out_batch[i] = (float)(i / M);
}

// ---------------------------------------------------------------------------
extern "C" void kernel_launch(void* const* d_in, const int* in_sizes, int n_in,
                              void* d_out, int out_size, void* d_ws, size_t ws_size,
                              hipStream_t stream) {
  (void)in_sizes; (void)n_in; (void)out_size; (void)ws_size;

  const float* fea = (const float*)d_in[0];   // [8,8192,3]
  const float* xyz = (const float*)d_in[1];   // [8,8192,3]
  // d_in[2] = batch (reconstructed analytically in finalize)

  const float* Wsrc[9]; const float* Bsrc[9];
  for (int l = 0; l < 9; ++l) {
    Wsrc[l] = (const float*)d_in[3 + 2 * l];
    Bsrc[l] = (const float*)d_in[4 + 2 * l];
  }
  static const int Kin[9]  = {6, 32, 32,   67, 96, 96,   131, 128, 128};
  static const int Kpad[9] = {8, 32, 32,   68, 96, 96,   132, 128, 128};
  static const int Dout[9] = {32, 32, 64,  96, 96, 128,  128, 128, 256};

  char* ws = (char*)d_ws;
  size_t off = 0;
  auto alloc = [&](size_t bytes) -> char* {
    char* p = ws + off;
    off = (off + bytes + 255) & ~(size_t)255;
    return p;
  };

  float* Wp[9];
  for (int l = 0; l < 9; ++l) Wp[l] = (float*)alloc(sizeof(float) * Kpad[l] * Dout[l]);
  int*   idx1 = (int*)  alloc(sizeof(int)   * BATCH * 2048);
  float* pos1 = (float*)alloc(sizeof(float) * BATCH * 2048 * 3);
  float* x1   = (float*)alloc(sizeof(float) * BATCH * 2048 * 64);
  int*   idx2 = (int*)  alloc(sizeof(int)   * BATCH * 1024);
  float* pos2 = (float*)alloc(sizeof(float) * BATCH * 1024 * 3);
  float* x2   = (float*)alloc(sizeof(float) * BATCH * 1024 * 128);
  int*   idx3 = (int*)  alloc(sizeof(int)   * BATCH * 512);
  float* pos3 = (float*)alloc(sizeof(float) * BATCH * 512 * 3);
  int*   nbr  = (int*)  alloc(sizeof(int)   * BATCH * 2048 * KNN_K);

  float* outF   = (float*)d_out;
  float* fea3   = outF;                       // [8,512,256]
  float* oxyz   = outF + BATCH * 512 * 256;   // [8,512,3]
  float* obatch = oxyz + BATCH * 512 * 3;     // [8,512]

  // K-pad all weights (cheap; deterministic each call).
  for (int l = 0; l < 9; ++l) {
    const int n = Kpad[l] * Dout[l];
    pad_weight_kernel<<<(n + 255) / 256, 256, 0, stream>>>(Wsrc[l], Wp[l], Kin[l], Kpad[l], Dout[l]);
  }

  // ---- SA1: 8192 -> 2048, MLP [6->32->32->64]
  fps_kernel<<<BATCH, FPS_THREADS, 0, stream>>>(xyz, idx1, pos1, NPTS, 2048);
  {
    const int total = BATCH * 2048;
    knn_kernel<<<(total + 255) / 256, 256, 0, stream>>>(xyz, pos1, nbr, NPTS, 2048, total);
    sa_mlp_kernel<<<total, MLP_THREADS, 0, stream>>>(
        fea, 3, xyz, pos1, nbr,
        Wp[0], Bsrc[0], 8, 32, Wp[1], Bsrc[1], 32, Wp[2], Bsrc[2], 64,
        x1, NPTS, 2048);
  }

  // ---- SA2: 2048 -> 1024, MLP [67->96->96->128]
  fps_kernel<<<BATCH, FPS_THREADS, 0, stream>>>(pos1, idx2, pos2, 2048, 1024);
  {
    const int total = BATCH * 1024;
    knn_kernel<<<(total + 255) / 256, 256, 0, stream>>>(pos1, pos2, nbr, 2048, 1024, total);
    sa_mlp_kernel<<<total, MLP_THREADS, 0, stream>>>(
        x1, 64, pos1, pos2, nbr,
        Wp[3], Bsrc[3], 68, 96, Wp[4], Bsrc[4], 96, Wp[5], Bsrc[5], 128,
        x2, 2048, 1024);
  }

  // ---- SA3: 1024 -> 512, MLP [131->128->128->256], writes fea3 directly.
  fps_kernel<<<BATCH, FPS_THREADS, 0, stream>>>(pos2, idx3, pos3, 1024, 512);
  {
    const int total = BATCH * 512;
    knn_kernel<<<(total + 255) / 256, 256, 0, stream>>>(pos2, pos3, nbr, 1024, 512, total);
    sa_mlp_kernel<<<total, MLP_THREADS, 0, stream>>>(
        x2, 128, pos2, pos3, nbr,
        Wp[6], Bsrc[6], 132, 128, Wp[7], Bsrc[7], 128, Wp[8], Bsrc[8], 256,
        fea3, 1024, 512);
  }

  finalize_kernel<<<(BATCH * 512 * 3 + 255) / 256, 256, 0, stream>>>(pos3, oxyz, obatch, 512);
}